// ClockworkRNN_8933531976483
// MI455X (gfx1250) — compile-verified
//
#include <hip/hip_runtime.h>

#define M_MOD   8
#define H_SZ    128
#define IN_SZ   512
#define MH      (M_MOD * H_SZ)     // 1024
#define T_STEPS 256
#define B_SZ    128

typedef __attribute__((ext_vector_type(2))) float v2f;
typedef __attribute__((ext_vector_type(8))) float v8f;

// D = A(16x4, f32) * B(4x16, f32) + C(16x16, f32)   -- native CDNA5 fp32 WMMA
__device__ __forceinline__ v8f wmma4_f32(v2f a, v2f b, v8f c) {
    return __builtin_amdgcn_wmma_f32_16x16x4_f32(
        /*neg_a=*/false, a, /*neg_b=*/false, b,
        /*c_mod=*/(short)0, c, /*reuse_a=*/false, /*reuse_b=*/false);
}

// ---------------------------------------------------------------------------
// Weight pre-swizzle into WMMA B-fragment quad order:
//   pk[((k>>2)*MH + n)*4 + (k&3)] = Bmat[k][n]
// so a lane's B fragment (two K rows of one column) is ONE contiguous b64 load
// and a half-wave covers a contiguous 256B chunk.
//   pk_whh: Bmat[k][n] = mask(n,k) ? Whh[n][k] : 0   (k over MH)
//   pk_wxh: Bmat[k][n] = Wxh[n][k]                   (k over IN)
// ---------------------------------------------------------------------------
__global__ void prep_weights(const float* __restrict__ Whh,
                             const float* __restrict__ Wxh,
                             float* __restrict__ pk_whh,
                             float* __restrict__ pk_wxh) {
    size_t idx = (size_t)blockIdx.x * blockDim.x + threadIdx.x;
    if (idx < (size_t)MH * MH) {
        int n = (int)(idx / MH);        // output row of Whh_m (module n/H)
        int k = (int)(idx % MH);        // contraction col
        float v = (k >= (n & ~(H_SZ - 1))) ? Whh[idx] : 0.0f;
        pk_whh[((size_t)(k >> 2) * MH + n) * 4 + (k & 3)] = v;
    }
    if (idx < (size_t)MH * IN_SZ) {
        int n = (int)(idx / IN_SZ);
        int k = (int)(idx % IN_SZ);
        pk_wxh[((size_t)(k >> 2) * MH + n) * 4 + (k & 3)] = Wxh[idx];
    }
}

// ---------------------------------------------------------------------------
// xW + bh for all timesteps:  out[tb, n] = sum_i x[tb,i] * Wxh[n,i] + bh[n]
// Block = 4 waves tiling M; block tile 128(M) x 64(N); wave tile 32(M) x 64(N)
// Per K-quad: 2 A b64 + 4 B b64 loads -> 8 WMMA.
// ---------------------------------------------------------------------------
__global__ __launch_bounds__(128)
void xw_kernel(const float* __restrict__ x,       // [T*B, IN]
               const float* __restrict__ pk_wxh,  // packed quads [IN/4][MH][4]
               const float* __restrict__ bh,      // [MH]
               float* __restrict__ out) {         // [T*B, MH]
    const int lane  = threadIdx.x & 31;
    const int wave  = threadIdx.x >> 5;
    const int lrow  = lane & 15;
    const int lhalf = lane >> 4;
    const int m0 = blockIdx.x * 128 + wave * 32;
    const int n0 = blockIdx.y * 64;

    v8f acc[2][4];
#pragma unroll
    for (int j = 0; j < 4; ++j) {
        float bv = bh[n0 + 16 * j + lrow];
#pragma unroll
        for (int i = 0; i < 2; ++i)
#pragma unroll
            for (int r = 0; r < 8; ++r) acc[i][j][r] = bv;
    }

    const float* a0ptr = x + (size_t)(m0 + lrow) * IN_SZ + 2 * lhalf;
    const float* a1ptr = a0ptr + (size_t)16 * IN_SZ;
    const float* bptr  = pk_wxh + (size_t)(n0 + lrow) * 4 + 2 * lhalf;

    for (int k4 = 0; k4 < IN_SZ / 4; ++k4) {
        v2f a0 = *(const v2f*)(a0ptr + 4 * k4);
        v2f a1 = *(const v2f*)(a1ptr + 4 * k4);
        const float* bk = bptr + (size_t)k4 * MH * 4;
#pragma unroll
        for (int j = 0; j < 4; ++j) {
            v2f b = *(const v2f*)(bk + 64 * j);      // 16 cols * 4 quad floats
            acc[0][j] = wmma4_f32(a0, b, acc[0][j]);
            acc[1][j] = wmma4_f32(a1, b, acc[1][j]);
        }
    }

#pragma unroll
    for (int i = 0; i < 2; ++i)
#pragma unroll
        for (int j = 0; j < 4; ++j)
#pragma unroll
            for (int r = 0; r < 8; ++r)
                out[(size_t)(m0 + 16 * i + r + 8 * lhalf) * MH + n0 + 16 * j + lrow] =
                    acc[i][j][r];
}

// ---------------------------------------------------------------------------
// One recurrent step.  On entry out_t holds xW_t + bh.
//   active cols (n < thr):  out_t[b,n] += sum_c hprev[b,c] * Whh_m[n,c]
//   inactive:               out_t[b,n]  = hprev[b,n]
// Grid: (B/16, MH/128).  Block = 2 waves; wave tile 16(M=batch) x 64(N=rows).
// Branch is block-uniform -> EXEC all-ones in the WMMA path.
// ---------------------------------------------------------------------------
__global__ __launch_bounds__(64)
void step_kernel(const float* __restrict__ hprev,   // [B, MH]
                 const float* __restrict__ pk_whh,  // packed quads [MH/4][MH][4]
                 float* __restrict__ out_t,         // [B, MH]
                 int thr) {
    const int m0    = blockIdx.x * 16;     // batch tile
    const int nbase = blockIdx.y * 128;    // hidden-row tile (thr multiple of 128)

    if (nbase >= thr) {
        // inactive modules: copy previous state, b128 moves
        for (int e = threadIdx.x; e < 16 * 32; e += 64) {
            int r  = e >> 5;
            int c  = nbase + (e & 31) * 4;
            *(float4*)&out_t[(size_t)(m0 + r) * MH + c] =
                *(const float4*)&hprev[(size_t)(m0 + r) * MH + c];
        }
        return;
    }

    const int lane  = threadIdx.x & 31;
    const int wave  = threadIdx.x >> 5;
    const int lrow  = lane & 15;
    const int lhalf = lane >> 4;
    const int n0 = nbase + wave * 64;

    v8f acc[4];
#pragma unroll
    for (int j = 0; j < 4; ++j)
#pragma unroll
        for (int r = 0; r < 8; ++r)
            acc[j][r] = out_t[(size_t)(m0 + r + 8 * lhalf) * MH + n0 + 16 * j + lrow];

    const float* aptr = hprev + (size_t)(m0 + lrow) * MH + 2 * lhalf;
    const float* bptr = pk_whh + (size_t)(n0 + lrow) * 4 + 2 * lhalf;

    for (int k4 = 0; k4 < MH / 4; ++k4) {
        v2f a = *(const v2f*)(aptr + 4 * k4);
        const float* bk = bptr + (size_t)k4 * MH * 4;
#pragma unroll
        for (int j = 0; j < 4; ++j) {
            v2f b = *(const v2f*)(bk + 64 * j);
            acc[j] = wmma4_f32(a, b, acc[j]);
        }
    }

#pragma unroll
    for (int j = 0; j < 4; ++j)
#pragma unroll
        for (int r = 0; r < 8; ++r)
            out_t[(size_t)(m0 + r + 8 * lhalf) * MH + n0 + 16 * j + lrow] = acc[j][r];
}

// ---------------------------------------------------------------------------
extern "C" void kernel_launch(void* const* d_in, const int* in_sizes, int n_in,
                              void* d_out, int out_size, void* d_ws, size_t ws_size,
                              hipStream_t stream) {
    (void)in_sizes; (void)n_in; (void)out_size; (void)ws_size;

    const float* x      = (const float*)d_in[0];   // [T, B, IN]
    const float* h_prev = (const float*)d_in[1];   // [B, MH]
    const float* Wxh    = (const float*)d_in[2];   // [MH, IN]
    const float* Whh    = (const float*)d_in[3];   // [MH, MH]
    const float* bh     = (const float*)d_in[4];   // [MH]
    float* out = (float*)d_out;                    // [T, B, MH] ++ [B, MH]

    float* pk_whh = (float*)d_ws;                  // MH*MH floats (4 MB)
    float* pk_wxh = pk_whh + (size_t)MH * MH;      // IN*MH floats (2 MB)

    // 1) mask + swizzle weights into WMMA fragment order
    {
        size_t n = (size_t)MH * MH;
        int blocks = (int)((n + 255) / 256);
        prep_weights<<<blocks, 256, 0, stream>>>(Whh, Wxh, pk_whh, pk_wxh);
    }

    // 2) xW + bh for all timesteps, written straight into d_out
    {
        dim3 grid(T_STEPS * B_SZ / 128, MH / 64);
        xw_kernel<<<grid, 128, 0, stream>>>(x, pk_wxh, bh, out);
    }

    // 3) sequential recurrent scan: 256 dependent small WMMA GEMMs
    for (int t = 0; t < T_STEPS; ++t) {
        int ma = 0;
        while (ma < M_MOD && (t % (1 << ma)) == 0) ma++;
        int thr = ma * H_SZ;

        const float* hp = (t == 0) ? h_prev : out + (size_t)(t - 1) * B_SZ * MH;
        float* ot = out + (size_t)t * B_SZ * MH;

        dim3 grid(B_SZ / 16, MH / 128);
        step_kernel<<<grid, 64, 0, stream>>>(hp, pk_whh, ot, thr);
    }

    // 4) h_last = out[T-1]
    hipMemcpyAsync(out + (size_t)T_STEPS * B_SZ * MH,
                   out + (size_t)(T_STEPS - 1) * B_SZ * MH,
                   (size_t)B_SZ * MH * sizeof(float),
                   hipMemcpyDeviceToDevice, stream);
}